// Decoder_11716670783827
// MI455X (gfx1250) — compile-verified
//
#include <hip/hip_runtime.h>
#include <math.h>

#define DD 128

typedef __attribute__((ext_vector_type(2))) float v2f;
typedef __attribute__((ext_vector_type(8))) float v8f;

// ---------------------------------------------------------------------------
// Kernel 1: segment boundaries.  batch_clu is sorted, so seg b spans
// [lower_bound(b), lower_bound(b+1)).  One thread per boundary (B+1 total).
// ---------------------------------------------------------------------------
__global__ void seg_bounds_kernel(const int* __restrict__ batch,
                                  int* __restrict__ segStart, int n, int nseg) {
  int t = blockIdx.x * blockDim.x + threadIdx.x;
  if (t > nseg) return;
  int lo = 0, hi = n;
  while (lo < hi) {
    int mid = (lo + hi) >> 1;
    if (batch[mid] < t) lo = mid + 1; else hi = mid;
  }
  segStart[t] = lo;
}

// ---------------------------------------------------------------------------
// Kernel 2: column means of dist_embedding [M,6] and dist_embedding_norm [M,1]
// ---------------------------------------------------------------------------
__global__ void means_kernel(const float* __restrict__ de,
                             const float* __restrict__ den,
                             float* __restrict__ means, int mrows) {
  int t = threadIdx.x;
  if (t < 6) {
    float s = 0.f;
    for (int r = 0; r < mrows; ++r) s += de[r * 6 + t];
    means[t] = s / (float)mrows;
  } else if (t == 6) {
    float s = 0.f;
    for (int r = 0; r < mrows; ++r) s += den[r];
    means[6] = s / (float)mrows;
  }
}

// ---------------------------------------------------------------------------
// Kernel 3: single-pass online-softmax attention pooling.
// One block (8 wave32) per segment.  Each wave streams whole rows of x
// (float4 per lane = 512B coalesced), butterfly-reduces the gate logit,
// and maintains (running max m, denom s, weighted-sum acc) with rescaling.
// Waves are merged through LDS at the end.  x_clu is read exactly once.
// ---------------------------------------------------------------------------
__global__ __launch_bounds__(256) void attn_pool_kernel(
    const float* __restrict__ x, const int* __restrict__ mask,
    const int* __restrict__ segStart,
    const float* __restrict__ gate_w, const float* __restrict__ gate_b,
    float* __restrict__ xs) {
  const int b    = blockIdx.x;
  const int tid  = threadIdx.x;
  const int lane = tid & 31;
  const int wave = tid >> 5;           // 0..7
  const int start = segStart[b];
  const int end   = segStart[b + 1];

  const float4 gwv = ((const float4*)gate_w)[lane];  // gate_w[lane*4 .. +3]
  const float  gb  = gate_b[0];

  float  m = -INFINITY, s = 0.f;
  float4 acc = make_float4(0.f, 0.f, 0.f, 0.f);

  for (int i = start + wave; i < end; i += 8) {
    if (!mask[i]) continue;                               // excluded entirely
    const float4 xv = ((const float4*)(x + (size_t)i * DD))[lane];
    float p = xv.x * gwv.x + xv.y * gwv.y + xv.z * gwv.z + xv.w * gwv.w;
#pragma unroll
    for (int off = 16; off; off >>= 1) p += __shfl_xor(p, off, 32);
    const float g = p + gb;                               // same in all lanes
    if (g > m) {
      const float sc = __expf(m - g);                     // exp(-inf)=0 first time
      s = s * sc + 1.f;
      acc.x = acc.x * sc + xv.x; acc.y = acc.y * sc + xv.y;
      acc.z = acc.z * sc + xv.z; acc.w = acc.w * sc + xv.w;
      m = g;
    } else {
      const float e = __expf(g - m);
      s += e;
      acc.x += e * xv.x; acc.y += e * xv.y;
      acc.z += e * xv.z; acc.w += e * xv.w;
    }
  }

  __shared__ float wm[8], wsum[8];
  __shared__ float wacc[8][DD];
  if (lane == 0) { wm[wave] = m; wsum[wave] = s; }
  wacc[wave][lane * 4 + 0] = acc.x;
  wacc[wave][lane * 4 + 1] = acc.y;
  wacc[wave][lane * 4 + 2] = acc.z;
  wacc[wave][lane * 4 + 3] = acc.w;
  __syncthreads();

  if (tid < DD) {
    float M = -INFINITY;
#pragma unroll
    for (int w = 0; w < 8; ++w) M = fmaxf(M, wm[w]);
    float denom = 0.f, num = 0.f;
    if (M > -INFINITY) {
#pragma unroll
      for (int w = 0; w < 8; ++w) {
        const float f = __expf(wm[w] - M);                // -inf wave -> 0
        denom += f * wsum[w];
        num   += f * wacc[w][tid];
      }
    }
    xs[(size_t)b * DD + tid] = num / fmaxf(denom, 1e-12f);  // matches reference
  }
}

// ---------------------------------------------------------------------------
// Kernel 4: fused MLPs in fp32 via V_WMMA_F32_16X16X4_F32.
// One block = 16 rows of x_scene.  8 waves each own one 16-wide column tile
// of w1/w1n (32 WMMAs over K=128), ReLU+bias into LDS (stride 132 = bank-
// conflict-free), then wave0/wave1 run the zero-padded 16x6 / 16x1 GEMMs and
// write (vel+b2)*mean_de / (nrm+b2n)*mean_den.
//
// Fragment layouts per CDNA5 ISA 7.12.2 (wave32):
//   A 16x4 f32 : v0 = {lanes0-15: K=k, lanes16-31: K=k+2}, v1 = K=k+1/k+3, M=lane%16
//   B 4x16 f32 : v0 = {lanes0-15: K=k, lanes16-31: K=k+2}, v1 = K=k+1/k+3, N=lane%16
//   C/D 16x16  : v[j] = row (j + 8*(lane>=16)), col = lane%16
// ---------------------------------------------------------------------------
__global__ __launch_bounds__(256) void mlp_kernel(
    const float* __restrict__ xs,
    const float* __restrict__ w1,  const float* __restrict__ b1,
    const float* __restrict__ w2,  const float* __restrict__ b2,
    const float* __restrict__ w1n, const float* __restrict__ b1n,
    const float* __restrict__ w2n, const float* __restrict__ b2n,
    const float* __restrict__ means,
    float* __restrict__ out, int normOff) {
  __shared__ float As[16 * 132];
  __shared__ float Hs[16 * 132];
  __shared__ float Hn[16 * 132];
  const int tid  = threadIdx.x;
  const int lane = tid & 31;
  const int wave = tid >> 5;
  const int m0   = blockIdx.x * 16;

  for (int idx = tid; idx < 16 * 128; idx += 256) {
    const int r = idx >> 7, c = idx & 127;
    As[r * 132 + c] = xs[(size_t)(m0 + r) * DD + c];
  }
  __syncthreads();

  const int row  = lane & 15;
  const int col  = lane & 15;
  const int half = lane >> 4;
  const int koff = half << 1;          // 0 or 2
  const int n0   = wave * 16;

  // ---- h = relu(A @ w1 + b1) -> Hs ----
  {
    v8f c8 = {};
    for (int k = 0; k < DD; k += 4) {
      v2f a; a.x = As[row * 132 + k + koff]; a.y = As[row * 132 + k + koff + 1];
      v2f bfr;
      bfr.x = w1[(k + koff) * DD + n0 + col];
      bfr.y = w1[(k + koff + 1) * DD + n0 + col];
      c8 = __builtin_amdgcn_wmma_f32_16x16x4_f32(false, a, false, bfr,
                                                 (short)0, c8, false, false);
    }
    const float bias = b1[n0 + col];
#pragma unroll
    for (int j = 0; j < 8; ++j)
      Hs[(j + half * 8) * 132 + n0 + col] = fmaxf(c8[j] + bias, 0.f);
  }
  // ---- hn = relu(A @ w1n + b1n) -> Hn ----
  {
    v8f c8 = {};
    for (int k = 0; k < DD; k += 4) {
      v2f a; a.x = As[row * 132 + k + koff]; a.y = As[row * 132 + k + koff + 1];
      v2f bfr;
      bfr.x = w1n[(k + koff) * DD + n0 + col];
      bfr.y = w1n[(k + koff + 1) * DD + n0 + col];
      c8 = __builtin_amdgcn_wmma_f32_16x16x4_f32(false, a, false, bfr,
                                                 (short)0, c8, false, false);
    }
    const float bias = b1n[n0 + col];
#pragma unroll
    for (int j = 0; j < 8; ++j)
      Hn[(j + half * 8) * 132 + n0 + col] = fmaxf(c8[j] + bias, 0.f);
  }
  __syncthreads();

  if (wave == 0) {               // vel = Hs @ w2 (6 cols, zero-padded to 16)
    v8f c8 = {};
    for (int k = 0; k < DD; k += 4) {
      v2f a; a.x = Hs[row * 132 + k + koff]; a.y = Hs[row * 132 + k + koff + 1];
      v2f bfr;
      if (col < 6) {
        bfr.x = w2[(k + koff) * 6 + col];
        bfr.y = w2[(k + koff + 1) * 6 + col];
      } else { bfr.x = 0.f; bfr.y = 0.f; }
      c8 = __builtin_amdgcn_wmma_f32_16x16x4_f32(false, a, false, bfr,
                                                 (short)0, c8, false, false);
    }
    if (col < 6) {
      const float bias = b2[col], mn = means[col];
#pragma unroll
      for (int j = 0; j < 8; ++j) {
        const int r = m0 + j + half * 8;
        out[(size_t)r * 6 + col] = (c8[j] + bias) * mn;
      }
    }
  } else if (wave == 1) {        // nrm = Hn @ w2n (1 col, zero-padded)
    v8f c8 = {};
    for (int k = 0; k < DD; k += 4) {
      v2f a; a.x = Hn[row * 132 + k + koff]; a.y = Hn[row * 132 + k + koff + 1];
      v2f bfr;
      if (col == 0) { bfr.x = w2n[k + koff]; bfr.y = w2n[k + koff + 1]; }
      else          { bfr.x = 0.f;           bfr.y = 0.f; }
      c8 = __builtin_amdgcn_wmma_f32_16x16x4_f32(false, a, false, bfr,
                                                 (short)0, c8, false, false);
    }
    if (col == 0) {
      const float bias = b2n[0], mn = means[6];
#pragma unroll
      for (int j = 0; j < 8; ++j) {
        const int r = m0 + j + half * 8;
        out[normOff + r] = (c8[j] + bias) * mn;
      }
    }
  }
}

// ---------------------------------------------------------------------------
extern "C" void kernel_launch(void* const* d_in, const int* in_sizes, int n_in,
                              void* d_out, int out_size, void* d_ws, size_t ws_size,
                              hipStream_t stream) {
  const float* x_clu = (const float*)d_in[0];
  const int*   mask  = (const int*)d_in[1];
  const int*   batch = (const int*)d_in[2];
  const float* de    = (const float*)d_in[3];
  const float* den   = (const float*)d_in[4];
  const float* gw    = (const float*)d_in[5];
  const float* gb    = (const float*)d_in[6];
  const float* w1    = (const float*)d_in[7];
  const float* b1    = (const float*)d_in[8];
  const float* w2    = (const float*)d_in[9];
  const float* b2    = (const float*)d_in[10];
  const float* w1n   = (const float*)d_in[11];
  const float* b1n   = (const float*)d_in[12];
  const float* w2n   = (const float*)d_in[13];
  const float* b2n   = (const float*)d_in[14];
  float* out = (float*)d_out;

  const int n     = in_sizes[2];          // N nodes
  const int nseg  = out_size / 7;         // B scenes (6+1 outputs per scene)
  const int mrows = in_sizes[3] / 6;      // M embedding rows

  // workspace: [0, 32KB) seg_start ints; [32KB) 7 mean floats; [64KB) x_scene
  int*   segStart = (int*)d_ws;
  float* meansPtr = (float*)d_ws + 8192;
  float* xs       = (float*)d_ws + 16384;

  seg_bounds_kernel<<<(nseg + 1 + 255) / 256, 256, 0, stream>>>(batch, segStart, n, nseg);
  means_kernel<<<1, 64, 0, stream>>>(de, den, meansPtr, mrows);
  attn_pool_kernel<<<nseg, 256, 0, stream>>>(x_clu, mask, segStart, gw, gb, xs);
  mlp_kernel<<<nseg / 16, 256, 0, stream>>>(xs, w1, b1, w2, b2, w1n, b1n,
                                            w2n, b2n, meansPtr, out, nseg * 6);
}